// MSDeformAttn_83107617178180
// MI455X (gfx1250) — compile-verified
//
#include <hip/hip_runtime.h>
#include <math.h>

typedef float v2f __attribute__((ext_vector_type(2)));
typedef float v8f __attribute__((ext_vector_type(8)));

#define S_TOT   21760
#define LQ      21760
#define DM      256
#define NH      8
#define HD      32

// ---------------------------------------------------------------------------
// One wave computes a 16x64 f32 C strip (4 x 16x16 WMMA tiles sharing the A
// fragment in registers): C[:, jbase:jbase+64] = A[mbase:mbase+16, :K] @ W^T.
// V_WMMA_F32_16X16X4_F32 fragment layout (wave32):
//   A 16x4 : lanes 0-15 rows m=lane hold K={k,k+1}; lanes 16-31 hold K={k+2,k+3}
//   B 4x16 : lanes 0-15 col n=lane hold K={k,k+1}; lanes 16-31 hold K={k+2,k+3}
//   C 16x16: VGPR v, lane: row = v + 8*(lane>>4), col = lane&15
// ---------------------------------------------------------------------------
__device__ __forceinline__ void wmma_gemm_strip64(const float* __restrict__ A,
                                                  const float* __restrict__ W,
                                                  int mbase, int jbase, int K,
                                                  v8f acc[4]) {
    const int lane = threadIdx.x & 31;
    const int half = lane >> 4;
    const int mn   = lane & 15;
    const float* __restrict__ arow = A + (size_t)(mbase + mn) * K + 2 * half;
    const float* __restrict__ br0  = W + (size_t)(jbase + mn) * K + 2 * half;
    const float* __restrict__ br1  = br0 + (size_t)16 * K;
    const float* __restrict__ br2  = br1 + (size_t)16 * K;
    const float* __restrict__ br3  = br2 + (size_t)16 * K;
    acc[0] = (v8f){0.f,0.f,0.f,0.f,0.f,0.f,0.f,0.f};
    acc[1] = acc[0]; acc[2] = acc[0]; acc[3] = acc[0];
#pragma unroll 4
    for (int k = 0; k < K; k += 4) {
        v2f a  = *(const v2f*)(arow + k);
        v2f b0 = *(const v2f*)(br0 + k);
        v2f b1 = *(const v2f*)(br1 + k);
        v2f b2 = *(const v2f*)(br2 + k);
        v2f b3 = *(const v2f*)(br3 + k);
        acc[0] = __builtin_amdgcn_wmma_f32_16x16x4_f32(false, a, false, b0,
                                                       (short)0, acc[0], false, false);
        acc[1] = __builtin_amdgcn_wmma_f32_16x16x4_f32(false, a, false, b1,
                                                       (short)0, acc[1], false, false);
        acc[2] = __builtin_amdgcn_wmma_f32_16x16x4_f32(false, a, false, b2,
                                                       (short)0, acc[2], false, false);
        acc[3] = __builtin_amdgcn_wmma_f32_16x16x4_f32(false, a, false, b3,
                                                       (short)0, acc[3], false, false);
    }
}

// value = input_flatten @ Wv^T + bv   -> (S, 256)
__global__ void msda_value_gemm(const float* __restrict__ inF,
                                const float* __restrict__ Wv,
                                const float* __restrict__ bv,
                                float* __restrict__ value) {
    const int wave  = (blockIdx.x << 3) + (threadIdx.x >> 5);
    const int mbase = (wave >> 2) << 4;   // 4 n-groups of 64 cols
    const int jbase = (wave & 3) << 6;
    v8f acc[4];
    wmma_gemm_strip64(inF, Wv, mbase, jbase, DM, acc);
    const int lane = threadIdx.x & 31;
    const int half = lane >> 4, n = lane & 15;
#pragma unroll
    for (int t = 0; t < 4; ++t) {
        const int col = jbase + t * 16 + n;
        const float bias = bv[col];
#pragma unroll
        for (int v = 0; v < 8; ++v) {
            const int row = mbase + v + 8 * half;
            value[(size_t)row * DM + col] = acc[t][v] + bias;
        }
    }
}

// loc = reference_points + (query @ Ws^T + bs) / normalizer  -> (Lq, 256)
// col j = h*32 + l*8 + p*2 + xy ; levels are square so norm depends only on l.
__global__ void msda_loc_gemm(const float* __restrict__ query,
                              const float* __restrict__ Wsamp,
                              const float* __restrict__ bs,
                              const float* __restrict__ refp,
                              float* __restrict__ loc) {
    const int wave  = (blockIdx.x << 3) + (threadIdx.x >> 5);
    const int mbase = (wave >> 2) << 4;
    const int jbase = (wave & 3) << 6;
    v8f acc[4];
    wmma_gemm_strip64(query, Wsamp, mbase, jbase, DM, acc);
    const int lane = threadIdx.x & 31;
    const int half = lane >> 4, n = lane & 15;
    const float dims[4] = {128.f, 64.f, 32.f, 16.f};
#pragma unroll
    for (int t = 0; t < 4; ++t) {
        const int col = jbase + t * 16 + n;
        const int l   = (col >> 3) & 3;
        const int xy  = col & 1;
        const float inv_norm = 1.0f / dims[l];
        const float bias = bs[col];
#pragma unroll
        for (int v = 0; v < 8; ++v) {
            const int row = mbase + v + 8 * half;
            const float ref = refp[(size_t)row * 8 + l * 2 + xy];
            loc[(size_t)row * DM + col] = ref + (acc[t][v] + bias) * inv_norm;
        }
    }
}

// aw_raw = query @ Wa^T + ba  -> (Lq, 128)
__global__ void msda_aw_gemm(const float* __restrict__ query,
                             const float* __restrict__ Wa,
                             const float* __restrict__ ba,
                             float* __restrict__ aw) {
    const int wave  = (blockIdx.x << 3) + (threadIdx.x >> 5);
    const int mbase = (wave >> 1) << 4;   // 2 n-groups of 64 cols (128 total)
    const int jbase = (wave & 1) << 6;
    v8f acc[4];
    wmma_gemm_strip64(query, Wa, mbase, jbase, DM, acc);
    const int lane = threadIdx.x & 31;
    const int half = lane >> 4, n = lane & 15;
#pragma unroll
    for (int t = 0; t < 4; ++t) {
        const int col = jbase + t * 16 + n;
        const float bias = ba[col];
#pragma unroll
        for (int v = 0; v < 8; ++v) {
            const int row = mbase + v + 8 * half;
            aw[(size_t)row * 128 + col] = acc[t][v] + bias;
        }
    }
}

// In-place softmax over the 16 (level,point) logits per (q, head).
__global__ void msda_softmax(float* __restrict__ aw) {
    const int t = blockIdx.x * blockDim.x + threadIdx.x;  // 21760*8 threads
    const int q = t >> 3, h = t & 7;
    float* __restrict__ p = aw + (size_t)q * 128 + h * 16;
    float vals[16], m = -1e30f;
#pragma unroll
    for (int i = 0; i < 16; ++i) { vals[i] = p[i]; m = fmaxf(m, vals[i]); }
    float s = 0.f;
#pragma unroll
    for (int i = 0; i < 16; ++i) { vals[i] = __expf(vals[i] - m); s += vals[i]; }
    const float inv = 1.0f / s;
#pragma unroll
    for (int i = 0; i < 16; ++i) p[i] = vals[i] * inv;
}

// Bilinear sampling + weighted sum. One wave per (q, head); lane = channel d.
// Each corner read is a contiguous 128B coalesced load from L2-resident value;
// all bilinear address math is wave-uniform (no divergence).
__global__ void msda_sample(const float* __restrict__ value,
                            const float* __restrict__ loc,
                            const float* __restrict__ aw,
                            float* __restrict__ heads) {
    const int w = (blockIdx.x << 3) + (threadIdx.x >> 5);
    const int q = w >> 3, h = w & 7;
    const int d = threadIdx.x & 31;
    const int lstart[4] = {0, 16384, 20480, 21504};
    const int ldim[4]   = {128, 64, 32, 16};
    const float* __restrict__ locq = loc + (size_t)q * DM + h * HD;
    const float* __restrict__ awq  = aw  + (size_t)q * 128 + h * 16;
    const int hoff = h * HD + d;
    float acc = 0.f;
#pragma unroll
    for (int l = 0; l < 4; ++l) {
        const int   Wd = ldim[l];
        const int   b0 = lstart[l];
        const float fW = (float)Wd;
#pragma unroll
        for (int p = 0; p < 4; ++p) {
            const float lx = locq[l * 8 + p * 2 + 0];
            const float ly = locq[l * 8 + p * 2 + 1];
            const float a  = awq[l * 4 + p];
            const float x = lx * fW - 0.5f;
            const float y = ly * fW - 0.5f;
            const float x0f = floorf(x), y0f = floorf(y);
            const float fx = x - x0f, fy = y - y0f;
            const int x0 = (int)x0f, y0 = (int)y0f;
            float s = 0.f;
            {   // (x0, y0)
                if (x0 >= 0 && x0 < Wd && y0 >= 0 && y0 < Wd)
                    s += (1.f - fx) * (1.f - fy) *
                         value[(size_t)(b0 + y0 * Wd + x0) * DM + hoff];
            }
            {   // (x0+1, y0)
                const int xi = x0 + 1;
                if (xi >= 0 && xi < Wd && y0 >= 0 && y0 < Wd)
                    s += fx * (1.f - fy) *
                         value[(size_t)(b0 + y0 * Wd + xi) * DM + hoff];
            }
            {   // (x0, y0+1)
                const int yi = y0 + 1;
                if (x0 >= 0 && x0 < Wd && yi >= 0 && yi < Wd)
                    s += (1.f - fx) * fy *
                         value[(size_t)(b0 + yi * Wd + x0) * DM + hoff];
            }
            {   // (x0+1, y0+1)
                const int xi = x0 + 1, yi = y0 + 1;
                if (xi >= 0 && xi < Wd && yi >= 0 && yi < Wd)
                    s += fx * fy *
                         value[(size_t)(b0 + yi * Wd + xi) * DM + hoff];
            }
            acc += a * s;
        }
    }
    heads[(size_t)q * DM + hoff] = acc;
}

// out = heads @ Wo^T + bo
__global__ void msda_out_gemm(const float* __restrict__ heads,
                              const float* __restrict__ Wo,
                              const float* __restrict__ bo,
                              float* __restrict__ out) {
    const int wave  = (blockIdx.x << 3) + (threadIdx.x >> 5);
    const int mbase = (wave >> 2) << 4;
    const int jbase = (wave & 3) << 6;
    v8f acc[4];
    wmma_gemm_strip64(heads, Wo, mbase, jbase, DM, acc);
    const int lane = threadIdx.x & 31;
    const int half = lane >> 4, n = lane & 15;
#pragma unroll
    for (int t = 0; t < 4; ++t) {
        const int col = jbase + t * 16 + n;
        const float bias = bo[col];
#pragma unroll
        for (int v = 0; v < 8; ++v) {
            const int row = mbase + v + 8 * half;
            out[(size_t)row * DM + col] = acc[t][v] + bias;
        }
    }
}

extern "C" void kernel_launch(void* const* d_in, const int* in_sizes, int n_in,
                              void* d_out, int out_size, void* d_ws, size_t ws_size,
                              hipStream_t stream) {
    const float* query = (const float*)d_in[0];
    const float* refp  = (const float*)d_in[1];
    const float* inF   = (const float*)d_in[2];
    const float* Wv    = (const float*)d_in[3];
    const float* bv    = (const float*)d_in[4];
    const float* Wsamp = (const float*)d_in[5];
    const float* bs    = (const float*)d_in[6];
    const float* Wa    = (const float*)d_in[7];
    const float* ba    = (const float*)d_in[8];
    const float* Wo    = (const float*)d_in[9];
    const float* bo    = (const float*)d_in[10];
    float* out = (float*)d_out;

    float* ws    = (float*)d_ws;
    float* value = ws;                                   // S*256
    float* loc   = value + (size_t)S_TOT * DM;           // Lq*256
    float* aw    = loc   + (size_t)LQ * DM;              // Lq*128
    float* heads = aw    + (size_t)LQ * 128;             // Lq*256

    const int MT = LQ / 16;                // 1360 M row-blocks
    const dim3 blk(256);
    // Phase 1: projections (independent; WMMA, 16x64 strip per wave)
    msda_value_gemm<<<dim3(MT * 4 / 8), blk, 0, stream>>>(inF, Wv, bv, value);
    msda_loc_gemm  <<<dim3(MT * 4 / 8), blk, 0, stream>>>(query, Wsamp, bs, refp, loc);
    msda_aw_gemm   <<<dim3(MT * 2 / 8), blk, 0, stream>>>(query, Wa, ba, aw);
    // Phase 2: softmax over 16 points per (q, head)
    msda_softmax   <<<dim3(LQ * NH / 256), blk, 0, stream>>>(aw);
    // Phase 3: bilinear gather + weighted accumulation
    msda_sample    <<<dim3(LQ), blk, 0, stream>>>(value, loc, aw, heads);
    // Phase 4: output projection
    msda_out_gemm  <<<dim3(MT * 4 / 8), blk, 0, stream>>>(heads, Wo, bo, out);
}